// ParametricDrumSynth_13400297964154
// MI455X (gfx1250) — compile-verified
//
#include <hip/hip_runtime.h>
#include <math.h>

#define SRATE     48000
#define NSAMP     480000
#define NUM_TONES 12
#define MAX_DEL   50
#define EPSV      1e-7f
#define TWO_PI_F  6.28318530717958647692f

#define CHUNK  1000
#define NCHUNK (NSAMP / CHUNK)   /* 480 chunks, exact */
#define WARM   4096              /* 0.99^4096 ~ 1e-18: exact to fp32 */

typedef __attribute__((ext_vector_type(2))) float v2f;
typedef __attribute__((ext_vector_type(8))) float v8f;

__device__ __forceinline__ float clipf(float x, float lo, float hi) {
    return fminf(fmaxf(x, lo), hi);
}

// ---------------------------------------------------------------------------
// Kernel 1: three one-pole IIR chains, parallelized by chunk + warm-up.
//   chain 0: filt_t = lowpass(noise_t, 0.1)
//   chain 1: filt_n = highpass(lowpass(noise_n, flp), fhp)
//   chain 2: filt_d = highpass(lowpass(delay_buffer, flo), fhi)
// ---------------------------------------------------------------------------
__global__ void drum_filters(const float* __restrict__ params,
                             const float* __restrict__ delay_buffer,
                             const float* __restrict__ noise_n,
                             const float* __restrict__ noise_t,
                             float* __restrict__ filt_t,
                             float* __restrict__ filt_n,
                             float* __restrict__ filt_d) {
    int tid = blockIdx.x * blockDim.x + threadIdx.x;
    if (tid >= 3 * NCHUNK) return;
    int chain = tid / NCHUNK;
    int chunk = tid - chain * NCHUNK;
    int start = chunk * CHUNK;
    int end   = start + CHUNK;
    int s0    = start - WARM; if (s0 < 0) s0 = 0;

    const float* __restrict__ x;
    float* __restrict__ y;
    float alo, ahi;
    int use_hp;
    if (chain == 0) {
        x = noise_t; y = filt_t;
        alo = 0.1f; ahi = 0.0f; use_hp = 0;
    } else if (chain == 1) {
        x = noise_n; y = filt_n;
        alo = clipf(fmaxf(params[48], EPSV), EPSV, 0.99f);
        ahi = clipf(fmaxf(params[49], EPSV), EPSV, 0.99f);
        use_hp = 1;
    } else {
        x = delay_buffer; y = filt_d;
        alo = clipf(fmaxf(params[43], EPSV), EPSV, 0.99f);
        ahi = clipf(fmaxf(params[44], EPSV), EPSV, 0.99f);
        use_hp = 1;
    }

    float one_m = 1.0f - alo;
    float lp = 0.0f, hp = 0.0f, lpprev = 0.0f;
    for (int n = s0; n < end; ++n) {
        float xv = x[n];
        lp = fmaf(alo, lp, one_m * xv);          // y = (1-a)x + a*y
        hp = ahi * (hp + lp - lpprev);           // y = a*(y_prev + x - x_prev)
        lpprev = lp;
        if (n >= start) y[n] = use_hp ? hp : lp;
    }
}

// ---------------------------------------------------------------------------
// Kernel 2: per-sample transient + tone bank + noise envelope.
// Writes mixed = (transient+tone)*0.5 and base = transient+tone+noise.
// ---------------------------------------------------------------------------
__global__ void drum_samples(const float* __restrict__ params,
                             const float* __restrict__ filt_t,
                             const float* __restrict__ filt_n,
                             float* __restrict__ mixed,
                             float* __restrict__ base) {
    int n = blockIdx.x * blockDim.x + threadIdx.x;
    if (n >= NSAMP) return;
    float t = (float)n * (1.0f / (float)(NSAMP - 1));

    // ---- transient (p[1..4]) ----
    float decay = fmaxf(params[1], EPSV);
    float freq  = fmaxf(fmaxf(params[2], EPSV), 1.0f);
    float sat   = clipf(fmaxf(params[3], EPSV), 0.01f, 100.0f);
    float gain  = fmaxf(params[4], EPSV);
    float sine  = sinf(TWO_PI_F * freq * t) * expf(-decay * t) * gain;
    sine = tanhf(sine * sat);
    float ndec  = clipf(decay * 100.0f, 0.1f, 1000.0f);
    float trns  = filt_t[n] * expf(-ndec * t) * (gain * 0.1f);
    float transient = sine + trns;

    // ---- tone bank (p[5..40]) ----
    float tone = 0.0f;
#pragma unroll
    for (int i = 0; i < NUM_TONES; ++i) {
        float f = fmaxf(fmaxf(params[5 + i],  EPSV), 1.0f);
        float d = fmaxf(params[17 + i], EPSV);
        float g = fmaxf(params[29 + i], EPSV);
        tone += sinf(TWO_PI_F * f * t) * expf(-d * t) * g;
    }

    // ---- noise (p[46..50]) ----
    float attack = fmaxf(params[46], EPSV);
    float ndecay = fmaxf(params[47], EPSV);
    float ngain  = fmaxf(params[50], EPSV);
    float env    = (1.0f - expf(-t * attack)) * expf(-t * ndecay);
    float noise  = filt_n[n] * env * ngain;

    mixed[n] = (transient + tone) * 0.5f;
    base[n]  = transient + tone + noise;
}

// ---------------------------------------------------------------------------
// Kernel 3: 49-tap geometric-weight FIR via V_WMMA_F32_16X16X4_F32.
// One wave per 16-output tile: A(16x4) = broadcast weights fb^i,
// B(4x16) = gathered filt_d columns, D accumulates -> every column of D
// holds the tap sum for output n0 + (lane&15). Then final tanh mix.
// ---------------------------------------------------------------------------
__global__ void drum_resonator(const float* __restrict__ params,
                               const float* __restrict__ filt_d,
                               const float* __restrict__ mixed,
                               const float* __restrict__ base,
                               float* __restrict__ out) {
    int lane = threadIdx.x & 31;
    int wave = blockIdx.x * (blockDim.x >> 5) + (threadIdx.x >> 5);
    int n0 = wave * 16;
    if (n0 >= NSAMP) return;          // wave-uniform guard (never splits EXEC)

    // resonator params p[41..45]
    float freq = clipf(fmaxf(params[41], EPSV), 20.0f, (float)SRATE * 0.5f);
    float fb   = clipf(fmaxf(params[42], EPSV), 0.0f, 0.99f);
    float gain = fmaxf(params[45], EPSV);
    int dl = (int)clipf(floorf((float)SRATE / freq), 1.0f, (float)(NSAMP / 2));
    int nd = NSAMP / dl; if (nd > MAX_DEL) nd = MAX_DEL;

    int j  = lane & 15;               // output column within tile
    int kb = (lane >> 4) << 1;        // k-pair this lane holds: {0,1} or {2,3}
    int n  = n0 + j;

    // weights: lane holds fb^(i0+kb) and fb^(i0+kb+1); advance by fb^4 per step
    float w0  = powf(fb, (float)(1 + kb));
    float fb4 = (fb * fb) * (fb * fb);

    v8f c = {};
#pragma unroll
    for (int cix = 0; cix < 13; ++cix) {          // taps i = 1..49 in 4-chunks
        int i0 = 1 + 4 * cix;
        int iA = i0 + kb;
        int iB = iA + 1;
        int sA = iA * dl;
        int sB = iB * dl;
        // mask guarantees (n - s) in [0, N) => reference's %N is identity
        bool vA = (iA < nd) && (n >= sA);
        bool vB = (iB < nd) && (n >= sB);
        int idxA = vA ? (n - sA) : 0;
        int idxB = vB ? (n - sB) : 0;
        float dA = filt_d[idxA]; dA = vA ? dA : 0.0f;
        float dB = filt_d[idxB]; dB = vB ? dB : 0.0f;

        v2f a; a.x = w0; a.y = w0 * fb;           // A[m][kb], A[m][kb+1]
        v2f b; b.x = dA; b.y = dB;                // B[kb][j], B[kb+1][j]
        c = __builtin_amdgcn_wmma_f32_16x16x4_f32(
                false, a, false, b, (short)0, c, false, false);
        w0 *= fb4;
    }

    // every D row of column j is identical -> c[0] = tap sum for sample n
    float taps = c[0];
    if (lane < 16) {
        float reso = tanhf((mixed[n] + taps) * gain);
        out[n] = tanhf(base[n] + reso);
    }
}

// ---------------------------------------------------------------------------
extern "C" void kernel_launch(void* const* d_in, const int* in_sizes, int n_in,
                              void* d_out, int out_size, void* d_ws, size_t ws_size,
                              hipStream_t stream) {
    const float* params       = (const float*)d_in[0];
    const float* delay_buffer = (const float*)d_in[1];
    const float* noise_n      = (const float*)d_in[2];
    const float* noise_t      = (const float*)d_in[3];
    float* out = (float*)d_out;

    float* ws     = (float*)d_ws;      // 5 * N floats = 9.6 MB scratch
    float* filt_t = ws + 0 * (size_t)NSAMP;
    float* filt_n = ws + 1 * (size_t)NSAMP;
    float* filt_d = ws + 2 * (size_t)NSAMP;
    float* mixedb = ws + 3 * (size_t)NSAMP;
    float* baseb  = ws + 4 * (size_t)NSAMP;

    int nfilt = 3 * NCHUNK;                                   // 1440 threads
    drum_filters<<<(nfilt + 127) / 128, 128, 0, stream>>>(
        params, delay_buffer, noise_n, noise_t, filt_t, filt_n, filt_d);

    drum_samples<<<(NSAMP + 255) / 256, 256, 0, stream>>>(
        params, filt_t, filt_n, mixedb, baseb);

    int waves = NSAMP / 16;                                   // 30000 tiles
    drum_resonator<<<waves / 8, 256, 0, stream>>>(            // 8 waves/block
        params, filt_d, mixedb, baseb, out);
}